// GNN_Encoder_32306744000893
// MI455X (gfx1250) — compile-verified
//
#include <hip/hip_runtime.h>

typedef float v2f __attribute__((ext_vector_type(2)));
typedef float v8f __attribute__((ext_vector_type(8)));

// ---------------------------------------------------------------- utilities
__global__ void k_zero(float* __restrict__ p, int n) {
    int i = blockIdx.x * blockDim.x + threadIdx.x;
    if (i < n) p[i] = 0.0f;
}

// degree of destinations (self-loop added later as +1)
__global__ void k_deg(const int* __restrict__ dst, float* __restrict__ deg, int nE) {
    int i = blockIdx.x * blockDim.x + threadIdx.x;
    if (i < nE) unsafeAtomicAdd(&deg[dst[i]], 1.0f);
}

// in-place deg -> dinv = rsqrt(deg + 1)
__global__ void k_dinv(float* __restrict__ d, int n) {
    int i = blockIdx.x * blockDim.x + threadIdx.x;
    if (i < n) d[i] = rsqrtf(d[i] + 1.0f);
}

// ------------------------------------------------- layer 1 GEMM: x[N,4]@W1[4,64]
// One wave handles 16 nodes. A tile is 16x4 f32 (2 VGPRs/lane), four 16-col
// B tiles cover the 64 outputs. Epilogue fuses *dinv[row] -> scaled1.
__global__ __launch_bounds__(128) void k_gemm1_wmma(
    const float* __restrict__ x, const float* __restrict__ W1,
    const float* __restrict__ dinv, float* __restrict__ s1, int n_nodes)
{
    int wave  = (blockIdx.x * blockDim.x + threadIdx.x) >> 5;
    int lane  = threadIdx.x & 31;
    int base  = wave * 16;
    if (base >= n_nodes) return;                 // wave-uniform exit (EXEC stays full)

    int mrow  = lane & 15;                       // A row within tile
    int khalf = lane >> 4;                       // 0: K=0,1  1: K=2,3
    int col   = lane & 15;                       // B/D column within tile

    // A: lane holds x[base+mrow][2*khalf], x[base+mrow][2*khalf+1]
    v2f a;
    a.x = x[(base + mrow) * 4 + 2 * khalf + 0];
    a.y = x[(base + mrow) * 4 + 2 * khalf + 1];

    // dinv for the 8 output rows this lane owns (M = v + 8*khalf)
    float dv[8];
#pragma unroll
    for (int v = 0; v < 8; ++v) dv[v] = dinv[base + 8 * khalf + v];

#pragma unroll
    for (int j = 0; j < 4; ++j) {                // 4 column tiles of 16
        v2f b;
        b.x = W1[(2 * khalf + 0) * 64 + j * 16 + col];
        b.y = W1[(2 * khalf + 1) * 64 + j * 16 + col];
        v8f c = {};
        c = __builtin_amdgcn_wmma_f32_16x16x4_f32(false, a, false, b,
                                                  (short)0, c, false, false);
#pragma unroll
        for (int v = 0; v < 8; ++v) {
            int m = 8 * khalf + v;
            s1[(size_t)(base + m) * 64 + j * 16 + col] = c[v] * dv[v];
        }
    }
}

// ------------------------------------------------- layer 2 GEMM: h1[N,64]@W2[64,32]
// K=64 as 16 chained 16x16x4 WMMAs; two 16-col accumulators cover 32 outputs.
__global__ __launch_bounds__(128) void k_gemm2_wmma(
    const float* __restrict__ h1, const float* __restrict__ W2,
    const float* __restrict__ dinv, float* __restrict__ s2, int n_nodes)
{
    int wave  = (blockIdx.x * blockDim.x + threadIdx.x) >> 5;
    int lane  = threadIdx.x & 31;
    int base  = wave * 16;
    if (base >= n_nodes) return;

    int mrow  = lane & 15;
    int khalf = lane >> 4;
    int col   = lane & 15;

    float dv[8];
#pragma unroll
    for (int v = 0; v < 8; ++v) dv[v] = dinv[base + 8 * khalf + v];

    v8f acc0 = {};
    v8f acc1 = {};
    const float* arow = h1 + (size_t)(base + mrow) * 64;

#pragma unroll
    for (int k = 0; k < 64; k += 4) {
        v2f a;
        a.x = arow[k + 2 * khalf + 0];
        a.y = arow[k + 2 * khalf + 1];
        v2f b0, b1;
        b0.x = W2[(k + 2 * khalf + 0) * 32 + col];
        b0.y = W2[(k + 2 * khalf + 1) * 32 + col];
        b1.x = W2[(k + 2 * khalf + 0) * 32 + 16 + col];
        b1.y = W2[(k + 2 * khalf + 1) * 32 + 16 + col];
        acc0 = __builtin_amdgcn_wmma_f32_16x16x4_f32(false, a, false, b0,
                                                     (short)0, acc0, false, false);
        acc1 = __builtin_amdgcn_wmma_f32_16x16x4_f32(false, a, false, b1,
                                                     (short)0, acc1, false, false);
    }

#pragma unroll
    for (int v = 0; v < 8; ++v) {
        int m = 8 * khalf + v;
        s2[(size_t)(base + m) * 32 + col]      = acc0[v] * dv[v];
        s2[(size_t)(base + m) * 32 + 16 + col] = acc1[v] * dv[v];
    }
}

// --------------------------------------------- edge scatter: agg[dst] += scaled[src]
// Each thread owns (edge, 4-float chunk): float4 gather + 4 HW f32 atomics.
template <int F>
__global__ void k_scatter(const int* __restrict__ src, const int* __restrict__ dst,
                          const float* __restrict__ s, float* __restrict__ agg, int nE)
{
    constexpr int CH = F >> 2;                   // chunks of 4 floats
    long long i = (long long)blockIdx.x * blockDim.x + threadIdx.x;
    long long total = (long long)nE * CH;
    if (i >= total) return;
    int e = (int)(i >> (F == 64 ? 4 : 3));
    int f = ((int)i & (CH - 1)) * 4;
    int sidx = src[e];
    int didx = dst[e];
    float4 v = *(const float4*)(s + (size_t)sidx * F + f);
    float* p = agg + (size_t)didx * F + f;
    unsafeAtomicAdd(p + 0, v.x);
    unsafeAtomicAdd(p + 1, v.y);
    unsafeAtomicAdd(p + 2, v.z);
    unsafeAtomicAdd(p + 3, v.w);
}

// ---------------------------- finalize layer 1 in place: agg1 -> h1 = relu(dinv*(agg+self)+b)
__global__ void k_finalize1(float* __restrict__ agg, const float* __restrict__ s1,
                            const float* __restrict__ dinv, const float* __restrict__ b,
                            int n_nodes)
{
    int i = blockIdx.x * blockDim.x + threadIdx.x;
    if (i >= n_nodes * 64) return;
    int n = i >> 6, f = i & 63;
    float h = dinv[n] * (agg[i] + s1[i]) + b[f];
    agg[i] = fmaxf(h, 0.0f);
}

// ---------------------------- finalize layer 2 + mean pool (LDS partial, atomic flush)
__global__ __launch_bounds__(256) void k_finalize2_pool(
    const float* __restrict__ agg2, const float* __restrict__ s2,
    const float* __restrict__ dinv, const float* __restrict__ b,
    float* __restrict__ pool, int n_nodes)
{
    __shared__ float sums[32];
    int i = blockIdx.x * 256 + threadIdx.x;
    int f = threadIdx.x & 31;
    if (threadIdx.x < 32) sums[threadIdx.x] = 0.0f;
    __syncthreads();
    float v = 0.0f;
    if (i < n_nodes * 32) {
        int n = i >> 5;
        v = fmaxf(dinv[n] * (agg2[i] + s2[i]) + b[f], 0.0f);
    }
    atomicAdd(&sums[f], v);                      // LDS atomic (ds_add_f32)
    __syncthreads();
    if (threadIdx.x < 32) unsafeAtomicAdd(&pool[threadIdx.x], sums[threadIdx.x]);
}

__global__ void k_mean(const float* __restrict__ pool, float* __restrict__ out, float invn) {
    int i = threadIdx.x;
    if (i < 32) out[i] = pool[i] * invn;
}

// ---------------------------------------------------------------- launcher
extern "C" void kernel_launch(void* const* d_in, const int* in_sizes, int n_in,
                              void* d_out, int out_size, void* d_ws, size_t ws_size,
                              hipStream_t stream) {
    const float* x  = (const float*)d_in[0];
    const int*   ei = (const int*)d_in[1];     // edge_index [2,E]
    const float* W1 = (const float*)d_in[2];
    const float* b1 = (const float*)d_in[3];
    const float* W2 = (const float*)d_in[4];
    const float* b2 = (const float*)d_in[5];
    float* out = (float*)d_out;

    const int N = in_sizes[0] / 4;
    const int E = in_sizes[1] / 2;
    const int* src = ei;
    const int* dst = ei + E;

    // workspace layout (floats): [pool 32][dinv N][agg1 N*64 -> h1][s1 N*64]
    float* ws   = (float*)d_ws;
    float* pool = ws;
    float* dinv = ws + 32;
    float* agg1 = dinv + N;
    float* s1   = agg1 + (size_t)N * 64;
    float* s2   = s1;                     // reuse after layer 1 (N*32)
    float* agg2 = s1 + (size_t)N * 32;    // second half of s1 region (N*32)

    const int TB = 256;
    // zero: pool + deg(dinv) + agg1 (contiguous prefix); s1 fully overwritten by gemm1
    {
        int cnt = 32 + N + N * 64;
        hipLaunchKernelGGL(k_zero, dim3((cnt + TB - 1) / TB), dim3(TB), 0, stream, ws, cnt);
    }
    hipLaunchKernelGGL(k_deg,  dim3((E + TB - 1) / TB), dim3(TB), 0, stream, dst, dinv, E);
    hipLaunchKernelGGL(k_dinv, dim3((N + TB - 1) / TB), dim3(TB), 0, stream, dinv, N);

    // layer 1: WMMA gemm (scale fused) -> scatter -> bias+relu
    {
        int waves  = (N + 15) / 16;
        int blocks = (waves + 3) / 4;     // 128 threads = 4 waves/block
        hipLaunchKernelGGL(k_gemm1_wmma, dim3(blocks), dim3(128), 0, stream,
                           x, W1, dinv, s1, N);
    }
    {
        long long total = (long long)E * 16;
        int blocks = (int)((total + TB - 1) / TB);
        hipLaunchKernelGGL(k_scatter<64>, dim3(blocks), dim3(TB), 0, stream,
                           src, dst, s1, agg1, E);
    }
    hipLaunchKernelGGL(k_finalize1, dim3((N * 64 + TB - 1) / TB), dim3(TB), 0, stream,
                       agg1, s1, dinv, b1, N);

    // zero agg2 region (aliases s1 second half; safe only after finalize1)
    hipLaunchKernelGGL(k_zero, dim3((N * 32 + TB - 1) / TB), dim3(TB), 0, stream,
                       agg2, N * 32);

    // layer 2: WMMA gemm (K=64) -> scatter -> finalize + pool
    {
        int waves  = (N + 15) / 16;
        int blocks = (waves + 3) / 4;
        hipLaunchKernelGGL(k_gemm2_wmma, dim3(blocks), dim3(128), 0, stream,
                           agg1 /*h1*/, W2, dinv, s2, N);
    }
    {
        long long total = (long long)E * 8;
        int blocks = (int)((total + TB - 1) / TB);
        hipLaunchKernelGGL(k_scatter<32>, dim3(blocks), dim3(TB), 0, stream,
                           src, dst, s2, agg2, E);
    }
    hipLaunchKernelGGL(k_finalize2_pool, dim3((N * 32 + 255) / 256), dim3(256), 0, stream,
                       agg2, s2, dinv, b2, pool, N);
    hipLaunchKernelGGL(k_mean, dim3(1), dim3(32), 0, stream, pool, out, 1.0f / (float)N);
}